// FinancialClaimGNN_75960791597188
// MI455X (gfx1250) — compile-verified
//
#include <hip/hip_runtime.h>

// ---------------------------------------------------------------------------
// FinancialClaimGNN on MI455X (gfx1250): hetero 2-layer SAGE + classifier.
// Scatter/gather phase = f32 atomics (node tables are L2-resident: ~115 MB of
// features vs 192 MB L2), GEMM phase = V_WMMA_F32_16X16X4_F32 with LDS-staged,
// fragment-swizzled 64x64 weights.
// ---------------------------------------------------------------------------

typedef __attribute__((ext_vector_type(2))) float v2f;
typedef __attribute__((ext_vector_type(8))) float v8f;

#define FDIM 64

// ---------------------------------------------------------------------------
// Edge scatter: one wave per edge; lane l adds cols {l, l+32} of the source
// row into agg[dst]. Optionally counts degree (lane 0). Prefetch next row.
// ---------------------------------------------------------------------------
__global__ void __launch_bounds__(256)
scatter_edges(const float* __restrict__ xsrc,
              const int*   __restrict__ src,
              const int*   __restrict__ dst,
              float*       __restrict__ agg,
              float*       __restrict__ cnt,
              int nedges, int do_cnt)
{
    const int lane  = threadIdx.x & 31;
    const int wave  = (int)(blockIdx.x * (blockDim.x >> 5) + (threadIdx.x >> 5));
    const int nwave = (int)(gridDim.x * (blockDim.x >> 5));

    for (int e = wave; e < nedges; e += nwave) {
        const int s = src[e];
        const int d = dst[e];
        const float* xp = xsrc + (long)s * FDIM;
        float*       ap = agg  + (long)d * FDIM;

        const int en = e + nwave;           // software prefetch of next source row
        if (en < nedges) {
            __builtin_prefetch(xsrc + (long)src[en] * FDIM, 0, 1);
        }

        atomicAdd(ap + lane,      xp[lane]);
        atomicAdd(ap + lane + 32, xp[lane + 32]);
        if (do_cnt && lane == 0) atomicAdd(cnt + d, 1.0f);
    }
}

// ---------------------------------------------------------------------------
// Fused SAGE GEMM for one destination node type (templated on edge-type count
// and activation so no runtime branches reach the WMMA inner loop):
//   out = act( mean0 @ W0  [+ mean1 @ W1]  +  X @ (Wr0 [+ Wr1])  + (b0 [+ b1]) )
// mean_i = A_i / max(cnt_i, 1) computed on the fly.
// One wave per 16-row tile; 4 accumulators of 16x16 f32; K=64 via 16 steps of
// V_WMMA_F32_16X16X4_F32. Weights staged in LDS in B-fragment order.
//
// Fragment layout (ISA 7.12.2, wave32):
//   A 16x4 : lane L, vgpr j -> A[L&15][k0 + 2*(L>>4) + j]   (contig float2)
//   B 4x16 : lane L, vgpr j -> W[k0 + j + 2*(L>>4)][n0 + (L&15)]
//   C 16x16: lane L, vgpr v -> row (v + 8*(L>>4)), col (L&15)
//
// LDS slots: 0 = W0 fragments, 1 = (Wr0 [+Wr1]) fragments, 2 = W1 fragments.
// ---------------------------------------------------------------------------
template <bool HASA1, bool RELU>
__global__ void __launch_bounds__(128)
sage_gemm(float*       __restrict__ out,
          const float* __restrict__ A0, const float* __restrict__ cnt0,
          const float* __restrict__ A1, const float* __restrict__ cnt1,
          const float* __restrict__ X,
          const float* __restrict__ W0, const float* __restrict__ W1,
          const float* __restrict__ Wr0, const float* __restrict__ Wr1,
          const float* __restrict__ b0, const float* __restrict__ b1,
          int n)
{
    __shared__ float wlds[3 * 4096];   // [slot][ks(16)][nt(4)][lane(32)][j(2)]

    const int tid = threadIdx.x;

    // Stage weight matrices as pre-swizzled B-fragments.
    const int nstage = (HASA1 ? 3 : 2) * 4096;
    for (int i = tid; i < nstage; i += blockDim.x) {
        const int j    = i & 1;
        const int lane = (i >> 1) & 31;
        const int nt   = (i >> 6) & 3;
        const int ks   = (i >> 8) & 15;
        const int m    = i >> 12;
        const int kk   = ks * 4 + ((lane >> 4) << 1) + j;
        const int nn   = nt * 16 + (lane & 15);
        float v;
        if (m == 0)      { v = W0[kk * FDIM + nn]; }
        else if (m == 1) { v = Wr0[kk * FDIM + nn];
                           if (HASA1) v += Wr1[kk * FDIM + nn]; }
        else             { v = W1[kk * FDIM + nn]; }
        wlds[i] = v;
    }
    __syncthreads();

    const int lane  = tid & 31;
    const int laneR = lane & 15;     // A row / C column within tile
    const int laneH = lane >> 4;     // which half of the wave
    const int wave  = (int)(blockIdx.x * (blockDim.x >> 5) + (tid >> 5));
    const int nwave = (int)(gridDim.x * (blockDim.x >> 5));
    const int ntile = n >> 4;        // node counts are multiples of 16

    // Bias per output column is tile-invariant: hoist out of the tile loop.
    float biasv[4];
    #pragma unroll
    for (int nt = 0; nt < 4; ++nt) {
        const int col = nt * 16 + laneR;
        biasv[nt] = b0[col];
        if (HASA1) biasv[nt] += b1[col];
    }

    for (int t = wave; t < ntile; t += nwave) {
        const long rowbase = (long)t << 4;
        const long arow    = rowbase + laneR;

        const float inv0 = 1.0f / fmaxf(cnt0[arow], 1.0f);
        float inv1 = 0.0f;
        if (HASA1) inv1 = 1.0f / fmaxf(cnt1[arow], 1.0f);

        const float* a0p = A0 + arow * FDIM + (laneH << 1);
        const float* xp  = X  + arow * FDIM + (laneH << 1);
        const float* a1p = HASA1 ? (A1 + arow * FDIM + (laneH << 1)) : nullptr;

        v8f acc[4];
        #pragma unroll
        for (int nt = 0; nt < 4; ++nt) acc[nt] = (v8f){0,0,0,0,0,0,0,0};

        // Partial unroll: keeps live A-fragments small so allocation stays
        // below VGPR 256 (avoids s_set_vgpr_msb thrash seen with full unroll).
        #pragma unroll 4
        for (int ks = 0; ks < 16; ++ks) {
            v2f fa0 = *(const v2f*)(a0p + ks * 4);
            fa0 *= inv0;
            v2f fx  = *(const v2f*)(xp + ks * 4);
            v2f fa1 = {0.0f, 0.0f};
            if (HASA1) { fa1 = *(const v2f*)(a1p + ks * 4); fa1 *= inv1; }

            #pragma unroll
            for (int nt = 0; nt < 4; ++nt) {
                const int fo = (ks * 4 + nt) * 64 + lane * 2;
                v2f bn = *(const v2f*)&wlds[fo];              // W0 fragment
                v2f br = *(const v2f*)&wlds[4096 + fo];       // combined Wr fragment
                acc[nt] = __builtin_amdgcn_wmma_f32_16x16x4_f32(
                              false, fa0, false, bn, (short)0, acc[nt], false, false);
                acc[nt] = __builtin_amdgcn_wmma_f32_16x16x4_f32(
                              false, fx,  false, br, (short)0, acc[nt], false, false);
                if (HASA1) {
                    v2f b1f = *(const v2f*)&wlds[2 * 4096 + fo];  // W1 fragment
                    acc[nt] = __builtin_amdgcn_wmma_f32_16x16x4_f32(
                                  false, fa1, false, b1f, (short)0, acc[nt], false, false);
                }
            }
        }

        // Epilogue: bias, optional ReLU, store (rows always in range).
        #pragma unroll
        for (int nt = 0; nt < 4; ++nt) {
            const int col = nt * 16 + laneR;
            #pragma unroll
            for (int v = 0; v < 8; ++v) {
                const long r = rowbase + v + (laneH << 3);
                float val = acc[nt][v] + biasv[nt];
                if (RELU) val = fmaxf(val, 0.0f);
                out[r * FDIM + col] = val;
            }
        }
    }
}

// ---------------------------------------------------------------------------
// Classifier: out[b, :] = h2[idx[b], :] @ Wc(64x2) + bc
// ---------------------------------------------------------------------------
__global__ void __launch_bounds__(256)
classify(const float* __restrict__ h2, const int* __restrict__ idx,
         const float* __restrict__ Wc, const float* __restrict__ bc,
         float* __restrict__ out, int B)
{
    const int b = (int)(blockIdx.x * blockDim.x + threadIdx.x);
    if (b >= B) return;
    const float* hp = h2 + (long)idx[b] * FDIM;
    float s0 = bc[0], s1 = bc[1];
    #pragma unroll 8
    for (int k = 0; k < FDIM; ++k) {
        const float h = hp[k];
        s0 += h * Wc[k * 2 + 0];
        s1 += h * Wc[k * 2 + 1];
    }
    out[(long)b * 2 + 0] = s0;
    out[(long)b * 2 + 1] = s1;
}

// ---------------------------------------------------------------------------
extern "C" void kernel_launch(void* const* d_in, const int* in_sizes, int n_in,
                              void* d_out, int out_size, void* d_ws, size_t ws_size,
                              hipStream_t stream)
{
    const float* x_claim = (const float*)d_in[0];
    const float* x_ent   = (const float*)d_in[1];
    const float* x_ev    = (const float*)d_in[2];
    const int*   ei      = (const int*)  d_in[3];   // [4, 2, E]
    const int*   bidx    = (const int*)  d_in[4];
    const float* Wl1     = (const float*)d_in[5];   // [4,64,64]
    const float* bl1     = (const float*)d_in[6];   // [4,64]
    const float* Wr1     = (const float*)d_in[7];
    const float* Wl2     = (const float*)d_in[8];
    const float* bl2     = (const float*)d_in[9];
    const float* Wr2     = (const float*)d_in[10];
    const float* Wc      = (const float*)d_in[11];  // [64,2]
    const float* bc      = (const float*)d_in[12];  // [2]
    float*       outp    = (float*)d_out;

    const long Nc = in_sizes[0] / FDIM;     // 100,000 claims
    const long Ne = in_sizes[1] / FDIM;     // 200,000 entities
    const long Nv = in_sizes[2] / FDIM;     // 150,000 evidence
    const int  E  = in_sizes[3] / 8;        // 2,000,000 edges per type
    const int  B  = in_sizes[4];            // 50,000

    // Workspace layout (floats). Counts + aggregation buffers up front (zeroed
    // together), then layer-1 hidden states. h2 reuses the entity agg region.
    float* ws  = (float*)d_ws;
    long   off = 0;
    float* c0  = ws + off; off += Nc;
    float* c1  = ws + off; off += Nc;
    float* c2  = ws + off; off += Ne;
    float* c3  = ws + off; off += Nv;
    float* a0  = ws + off; off += Nc * FDIM;
    float* a1  = ws + off; off += Nc * FDIM;
    float* a2  = ws + off; off += Ne * FDIM;
    float* a3  = ws + off; off += Nv * FDIM;
    const long zero_floats = off;
    float* h1c = ws + off; off += Nc * FDIM;
    float* h1e = ws + off; off += Ne * FDIM;
    float* h1v = ws + off; off += Nv * FDIM;
    float* h2c = a2;       // safe reuse: a2 dead after layer-1 entity GEMM

    // Edge arrays per type t: src = ei[t][0], dst = ei[t][1]
    const int* s0 = ei + (long)0 * 2 * E; const int* d0 = s0 + E;
    const int* s1 = ei + (long)1 * 2 * E; const int* d1 = s1 + E;
    const int* s2 = ei + (long)2 * 2 * E; const int* d2 = s2 + E;
    const int* s3 = ei + (long)3 * 2 * E; const int* d3 = s3 + E;

    hipMemsetAsync(d_ws, 0, (size_t)zero_floats * sizeof(float), stream);

    const dim3 sblk(256);
    const dim3 sgrd(8192);     // 65,536 waves; ~31 edges each

    // ---- Layer 1 aggregation (with degree counts) ----
    scatter_edges<<<sgrd, sblk, 0, stream>>>(x_ent,   s0, d0, a0, c0, E, 1); // entity  -> claim
    scatter_edges<<<sgrd, sblk, 0, stream>>>(x_ev,    s1, d1, a1, c1, E, 1); // evidence-> claim
    scatter_edges<<<sgrd, sblk, 0, stream>>>(x_claim, s2, d2, a2, c2, E, 1); // claim   -> entity
    scatter_edges<<<sgrd, sblk, 0, stream>>>(x_claim, s3, d3, a3, c3, E, 1); // claim   -> evidence

    auto ggrid = [](long n) { return dim3((unsigned)((n / 16 + 3) / 4)); };

    // ---- Layer 1 GEMMs (ReLU) ----
    sage_gemm<true, true><<<ggrid(Nc), 128, 0, stream>>>(h1c, a0, c0, a1, c1, x_claim,
        Wl1 + 0 * 4096, Wl1 + 1 * 4096, Wr1 + 0 * 4096, Wr1 + 1 * 4096,
        bl1 + 0 * 64,   bl1 + 1 * 64,   (int)Nc);
    sage_gemm<false, true><<<ggrid(Ne), 128, 0, stream>>>(h1e, a2, c2, nullptr, nullptr, x_ent,
        Wl1 + 2 * 4096, nullptr, Wr1 + 2 * 4096, nullptr,
        bl1 + 2 * 64,   nullptr, (int)Ne);
    sage_gemm<false, true><<<ggrid(Nv), 128, 0, stream>>>(h1v, a3, c3, nullptr, nullptr, x_ev,
        Wl1 + 3 * 4096, nullptr, Wr1 + 3 * 4096, nullptr,
        bl1 + 3 * 64,   nullptr, (int)Nv);

    // ---- Layer 2: only claim outputs are consumed. Reuse a0/a1 (contiguous). ----
    hipMemsetAsync(a0, 0, (size_t)Nc * FDIM * 2 * sizeof(float), stream);
    scatter_edges<<<sgrd, sblk, 0, stream>>>(h1e, s0, d0, a0, nullptr, E, 0);
    scatter_edges<<<sgrd, sblk, 0, stream>>>(h1v, s1, d1, a1, nullptr, E, 0);

    sage_gemm<true, false><<<ggrid(Nc), 128, 0, stream>>>(h2c, a0, c0, a1, c1, h1c,
        Wl2 + 0 * 4096, Wl2 + 1 * 4096, Wr2 + 0 * 4096, Wr2 + 1 * 4096,
        bl2 + 0 * 64,   bl2 + 1 * 64,   (int)Nc);

    // ---- Classifier ----
    classify<<<dim3((B + 255) / 256), dim3(256), 0, stream>>>(h2c, bidx, Wc, bc, outp, B);
}